// PolyclassFTTlayer_seqOld_54004918780531
// MI455X (gfx1250) — compile-verified
//
#include <hip/hip_runtime.h>

typedef __attribute__((ext_vector_type(2))) float v2f;
typedef __attribute__((ext_vector_type(8))) float v8f;

#define B_BATCH 256
#define S_DIM   16384
#define R_RANK  16
#define KSPLIT  8
#define NT_TOTAL 65            // 64 N-tiles for cores1..4 (16 each) + 1 tile for core5
#define NCOLS   1040           // 4*256 + 16 columns of V per batch
#define MBLK    4              // 4 M-tiles (64 batch rows) per wave
#define KCHUNK  (S_DIM / KSPLIT)   // 2048

__device__ __forceinline__ v8f wmma_f32x4(v2f a, v2f b, v8f c) {
    // (neg_a, A, neg_b, B, c_mod, C, reuse_a, reuse_b)
    return __builtin_amdgcn_wmma_f32_16x16x4_f32(false, a, false, b, (short)0, c,
                                                 false, false);
}

// ---------------------------------------------------------------------------
// Phase 1: V[b, col] = sum_s z[b,s] * core_col[s]   (GEMM 256 x 1040 x 16384)
// One wave computes a 64(batch) x 16(col) tile over a 2048-wide K slice.
// All hot-loop addresses are pointer bumps + immediate offsets.
// ---------------------------------------------------------------------------
__global__ __launch_bounds__(32)
void tt_phase1(const float* __restrict__ z,
               const float* __restrict__ c1, const float* __restrict__ c2,
               const float* __restrict__ c3, const float* __restrict__ c4,
               const float* __restrict__ c5,
               float* __restrict__ vpart) {
    const int lane = threadIdx.x & 31;
    const int half = lane >> 4;      // selects K pair within fragment
    const int l    = lane & 15;      // M (A) or N (B) index

    const int t  = blockIdx.x;       // N tile 0..64
    const int mb = blockIdx.y;       // 0..3  (64 batches each)
    const int ks = blockIdx.z;       // 0..KSPLIT-1

    const int b0      = mb * 64;
    const int k_begin = ks * KCHUNK;

    // Per-lane A row pointers, pre-offset by this lane's K pair.
    const float* ap0 = z + (size_t)(b0 + l) * S_DIM + k_begin + 2 * half;
    const float* ap1 = ap0 + (size_t)16 * S_DIM;
    const float* ap2 = ap0 + (size_t)32 * S_DIM;
    const float* ap3 = ap0 + (size_t)48 * S_DIM;

    v8f acc0 = {}, acc1 = {}, acc2 = {}, acc3 = {};
    int colbase;

    if (t < 64) {
        // cores 1..4, shape (16, S, 16): B(s, e) = cp[d*S*16 + s*16 + e]
        const int k = t >> 4;
        const int d = t & 15;
        const float* cp = (k == 0) ? c1 : (k == 1) ? c2 : (k == 2) ? c3 : c4;
        colbase = k * 256 + d * 16;
        // lane's B pointer: element (k_begin + 2*half, l)
        const float* bp = cp + (size_t)d * S_DIM * R_RANK
                             + (size_t)(k_begin + 2 * half) * R_RANK + l;
        for (int it = 0; it < KCHUNK / 16; ++it) {
            __builtin_prefetch(bp + 4 * 256, 0, 1);   // 4 iterations ahead
            __builtin_prefetch(ap0 + 64, 0, 1);
#pragma unroll
            for (int u = 0; u < 4; ++u) {
                v2f bf;
                bf.x = bp[u * 64];            // row (.. + u*4 + 2*half)
                bf.y = bp[u * 64 + 16];       // row (.. + u*4 + 2*half + 1)
                v2f a0 = *(const v2f*)(ap0 + u * 4);
                v2f a1 = *(const v2f*)(ap1 + u * 4);
                v2f a2 = *(const v2f*)(ap2 + u * 4);
                v2f a3 = *(const v2f*)(ap3 + u * 4);
                acc0 = wmma_f32x4(a0, bf, acc0);
                acc1 = wmma_f32x4(a1, bf, acc1);
                acc2 = wmma_f32x4(a2, bf, acc2);
                acc3 = wmma_f32x4(a3, bf, acc3);
            }
            ap0 += 16; ap1 += 16; ap2 += 16; ap3 += 16;
            bp  += 256;                        // 16 K rows * 16 floats
        }
    } else {
        // core5, shape (16, S, 1): B(s, d) = c5[d*S + s]  (contiguous in s)
        colbase = 1024;
        const float* bp = c5 + (size_t)l * S_DIM + k_begin + 2 * half;
        for (int it = 0; it < KCHUNK / 16; ++it) {
            __builtin_prefetch(bp + 64, 0, 1);
            __builtin_prefetch(ap0 + 64, 0, 1);
#pragma unroll
            for (int u = 0; u < 4; ++u) {
                v2f bf = *(const v2f*)(bp + u * 4);
                v2f a0 = *(const v2f*)(ap0 + u * 4);
                v2f a1 = *(const v2f*)(ap1 + u * 4);
                v2f a2 = *(const v2f*)(ap2 + u * 4);
                v2f a3 = *(const v2f*)(ap3 + u * 4);
                acc0 = wmma_f32x4(a0, bf, acc0);
                acc1 = wmma_f32x4(a1, bf, acc1);
                acc2 = wmma_f32x4(a2, bf, acc2);
                acc3 = wmma_f32x4(a3, bf, acc3);
            }
            ap0 += 16; ap1 += 16; ap2 += 16; ap3 += 16;
            bp  += 16;
        }
    }

    // D layout: VGPR v, lane: M = v + 8*half, N = l
    float* vp = vpart + (size_t)ks * B_BATCH * NCOLS
                      + (size_t)(b0 + 8 * half) * NCOLS + colbase + l;
    v8f accs[MBLK] = {acc0, acc1, acc2, acc3};
#pragma unroll
    for (int m = 0; m < MBLK; ++m) {
#pragma unroll
        for (int v = 0; v < 8; ++v) {
            vp[(size_t)(m * 16 + v) * NCOLS] = accs[m][v];
        }
    }
}

// ---------------------------------------------------------------------------
// Phase 2: reduce K-split partials and fold the R x R chain:
//   w[b] = V1[b] @ V2[b] @ V3[b] @ V4[b] @ v5[b]     (right-to-left matvecs)
// ---------------------------------------------------------------------------
__global__ __launch_bounds__(256)
void tt_phase2(const float* __restrict__ vpart, float* __restrict__ w) {
    const int b = blockIdx.x * blockDim.x + threadIdx.x;
    if (b >= B_BATCH) return;

    const size_t rowoff = (size_t)b * NCOLS;
    float t[R_RANK], nt[R_RANK];

#pragma unroll
    for (int d = 0; d < R_RANK; ++d) {       // v5 vector (cols 1024..1039)
        float s = 0.f;
        for (int p = 0; p < KSPLIT; ++p)
            s += vpart[(size_t)p * B_BATCH * NCOLS + rowoff + 1024 + d];
        t[d] = s;
    }
    for (int k = 3; k >= 0; --k) {           // V4, V3, V2, V1
#pragma unroll
        for (int d = 0; d < R_RANK; ++d) {
            float s = 0.f;
#pragma unroll
            for (int e = 0; e < R_RANK; ++e) {
                float v = 0.f;
                for (int p = 0; p < KSPLIT; ++p)
                    v += vpart[(size_t)p * B_BATCH * NCOLS + rowoff +
                               k * 256 + d * 16 + e];
                s += v * t[e];
            }
            nt[d] = s;
        }
#pragma unroll
        for (int d = 0; d < R_RANK; ++d) t[d] = nt[d];
    }
#pragma unroll
    for (int r = 0; r < R_RANK; ++r)
        w[b * R_RANK + r] = t[r];
}

// ---------------------------------------------------------------------------
// Phase 3: out[b, s] = sum_r core0[s, r] * w[b, r]   (rank-16 outer product)
// One wave computes a 16(batch) x 16(pixel) output tile with 4 WMMAs (K=16).
// ---------------------------------------------------------------------------
__global__ __launch_bounds__(32)
void tt_phase3(const float* __restrict__ w, const float* __restrict__ core0,
               float* __restrict__ out) {
    const int lane = threadIdx.x & 31;
    const int half = lane >> 4;
    const int l    = lane & 15;
    const int s0   = blockIdx.x * 16;
    const int b0   = blockIdx.y * 16;

    const float* wp = w     + (size_t)(b0 + l) * R_RANK + 2 * half;
    const float* cp = core0 + (size_t)(s0 + l) * R_RANK + 2 * half;

    v8f acc = {};
#pragma unroll
    for (int k0 = 0; k0 < R_RANK; k0 += 4) {
        v2f a  = *(const v2f*)(wp + k0);
        v2f bf = *(const v2f*)(cp + k0);
        acc = wmma_f32x4(a, bf, acc);
    }
    float* op = out + (size_t)(b0 + 8 * half) * S_DIM + s0 + l;
#pragma unroll
    for (int v = 0; v < 8; ++v) {
        op[(size_t)v * S_DIM] = acc[v];
    }
}

// ---------------------------------------------------------------------------
extern "C" void kernel_launch(void* const* d_in, const int* in_sizes, int n_in,
                              void* d_out, int out_size, void* d_ws, size_t ws_size,
                              hipStream_t stream) {
    const float* z  = (const float*)d_in[0];
    const float* c0 = (const float*)d_in[1];
    const float* c1 = (const float*)d_in[2];
    const float* c2 = (const float*)d_in[3];
    const float* c3 = (const float*)d_in[4];
    const float* c4 = (const float*)d_in[5];
    const float* c5 = (const float*)d_in[6];
    float* out = (float*)d_out;

    float* vpart = (float*)d_ws;                                  // KSPLIT*B*NCOLS f32
    float* w     = vpart + (size_t)KSPLIT * B_BATCH * NCOLS;      // B*R f32

    tt_phase1<<<dim3(NT_TOTAL, B_BATCH / 64, KSPLIT), 32, 0, stream>>>(
        z, c1, c2, c3, c4, c5, vpart);
    tt_phase2<<<dim3(1), 256, 0, stream>>>(vpart, w);
    tt_phase3<<<dim3(S_DIM / 16, B_BATCH / 16), 32, 0, stream>>>(w, c0, out);
}